// DenseLayer_36850819400348
// MI455X (gfx1250) — compile-verified
//
#include <hip/hip_runtime.h>

typedef __attribute__((ext_vector_type(2))) float v2f;
typedef __attribute__((ext_vector_type(8))) float v8f;

#define N_NODES 100000
#define N_EDGES 3200000
#define F_IN 256
#define H1 128
#define H2 32
#define BN_EPS 1e-5f

// ---------------- small utility kernels ----------------

__global__ void zero_f32(float* __restrict__ p, int n) {
  int i = blockIdx.x * blockDim.x + threadIdx.x;
  if (i < n) p[i] = 0.f;
}

__global__ void copy_f4(float4* __restrict__ dst, const float4* __restrict__ src, int n4) {
  int i = blockIdx.x * blockDim.x + threadIdx.x;
  if (i < n4) dst[i] = src[i];
}

// GIN aggregation: agg[dst[e]][f] += x[src[e]][f]   (agg pre-initialized to x)
// Mapping: consecutive threads -> same edge, consecutive features => coalesced
// gather loads and coalesced fire-and-forget global_atomic_add_f32.
__global__ void scatter_add(const float* __restrict__ x,
                            const int* __restrict__ srcIdx,
                            const int* __restrict__ dstIdx,
                            float* __restrict__ agg,
                            int nElems, int F) {
  int i = blockIdx.x * blockDim.x + threadIdx.x;
  if (i >= nElems) return;
  int e = i / F;
  int f = i - e * F;
  int s = srcIdx[e];
  int d = dstIdx[e];
  atomicAdd(&agg[d * F + f], x[s * F + f]);
}

// ---------------- fp32 WMMA GEMM: out = A[MxK] @ W[KxF] + bias, optional ReLU,
// optional per-channel (sum, sum^2) accumulation for BatchNorm batch stats ----
// One wave32 computes one 16x16 output tile via V_WMMA_F32_16X16X4_F32.
__global__ void gemm_wmma_f32(const float* __restrict__ A,
                              const float* __restrict__ W,
                              const float* __restrict__ bias,
                              float* __restrict__ out,
                              int M, int K, int F, int doRelu,
                              float* __restrict__ sSum,
                              float* __restrict__ sSq) {
  const int wavesPerBlock = blockDim.x >> 5;
  const int waveId = blockIdx.x * wavesPerBlock + (threadIdx.x >> 5);
  const int tilesN = F >> 4;
  const int tilesM = M >> 4;
  if (waveId >= tilesM * tilesN) return;  // wave-uniform: EXEC stays all-ones
  const int tm = waveId / tilesN;
  const int tn = waveId - tm * tilesN;
  const int m0 = tm << 4;
  const int n0 = tn << 4;
  const int lane = threadIdx.x & 31;
  const int half = lane >> 4;   // 0: K pair {0,1} / rows M, 1: K pair {2,3} / rows M+8
  const int l16 = lane & 15;

  v8f acc = {0.f, 0.f, 0.f, 0.f, 0.f, 0.f, 0.f, 0.f};

  // A lane base: row (m0+l16), starting at K offset 2*half (8B aligned -> float2 load)
  const float* aRow = A + (m0 + l16) * K + 2 * half;
  // B lane base: column (n0+l16), rows k+2*half and k+2*half+1
  const float* wCol = W + (2 * half) * F + n0 + l16;

  for (int k = 0; k < K; k += 4) {
    v2f a = *(const v2f*)(aRow + k);        // A[m][k+2h], A[m][k+2h+1]
    v2f b;
    b.x = wCol[k * F];                      // W[k+2h][n]
    b.y = wCol[k * F + F];                  // W[k+2h+1][n]
    acc = __builtin_amdgcn_wmma_f32_16x16x4_f32(
        /*neg_a=*/false, a, /*neg_b=*/false, b,
        /*c_mod=*/(short)0, acc, /*reuse_a=*/false, /*reuse_b=*/false);
  }

  const float bn = bias[n0 + l16];
  float s = 0.f, q = 0.f;
  float vals[8];
#pragma unroll
  for (int r = 0; r < 8; ++r) {
    float v = acc[r] + bn;
    if (doRelu) v = fmaxf(v, 0.f);
    vals[r] = v;
    s += v;
    q += v * v;
  }
#pragma unroll
  for (int r = 0; r < 8; ++r) {
    // D VGPR r holds row m0 + r + 8*half, column n0 + l16
    out[(m0 + r + 8 * half) * F + n0 + l16] = vals[r];
  }

  if (sSum != nullptr) {
    // lanes l and l+16 hold the same output column -> combine, then 16 atomics/wave
    s += __shfl_xor(s, 16, 32);
    q += __shfl_xor(q, 16, 32);
    if (lane < 16) {
      atomicAdd(&sSum[n0 + l16], s);
      atomicAdd(&sSq[n0 + l16], q);
    }
  }
}

// BatchNorm (training batch stats, biased var) + ReLU
__global__ void bn_relu(const float* __restrict__ t,
                        const float* __restrict__ sSum,
                        const float* __restrict__ sSq,
                        const float* __restrict__ gamma,
                        const float* __restrict__ beta,
                        float* __restrict__ out,
                        int total, int F, float invN) {
  int i = blockIdx.x * blockDim.x + threadIdx.x;
  if (i >= total) return;
  int f = i & (F - 1);  // F is a power of two (128 or 32)
  float mean = sSum[f] * invN;
  float var = sSq[f] * invN - mean * mean;
  float v = (t[i] - mean) * rsqrtf(var + BN_EPS) * gamma[f] + beta[f];
  out[i] = fmaxf(v, 0.f);
}

// ---------------- driver ----------------

extern "C" void kernel_launch(void* const* d_in, const int* in_sizes, int n_in,
                              void* d_out, int out_size, void* d_ws, size_t ws_size,
                              hipStream_t stream) {
  (void)in_sizes; (void)n_in; (void)out_size; (void)ws_size;

  const float* x   = (const float*)d_in[0];
  const int*   ei  = (const int*)d_in[1];   // edge_index [2, E], int32 on device
  const float* W1a = (const float*)d_in[2];
  const float* b1a = (const float*)d_in[3];
  const float* W1b = (const float*)d_in[4];
  const float* b1b = (const float*)d_in[5];
  const float* g1  = (const float*)d_in[6];
  const float* be1 = (const float*)d_in[7];
  const float* W2a = (const float*)d_in[8];
  const float* b2a = (const float*)d_in[9];
  const float* W2b = (const float*)d_in[10];
  const float* b2b = (const float*)d_in[11];
  const float* g2  = (const float*)d_in[12];
  const float* be2 = (const float*)d_in[13];

  const int* srcIdx = ei;
  const int* dstIdx = ei + N_EDGES;

  // workspace layout (floats)
  float* ws    = (float*)d_ws;
  float* hsum1 = ws;                    // 25,600,000 floats (x + agg, layer 1)
  float* t1    = ws + 25600000;         // 12,800,000 (relu(hsum1@W1a+b1a)); later h1
  float* t2    = ws + 38400000;         // 12,800,000 (t1@W1b+b1b)
  float* stats = ws + 51200000;         // 512 floats
  float* ssum1 = stats;                 // 128
  float* ssq1  = stats + 128;           // 128
  float* ssum2 = stats + 256;           // 32
  float* ssq2  = stats + 288;           // 32
  float* h1    = t1;                    // reuse: BN1 output overwrites t1
  float* hsum2 = hsum1;                 // reuse: 12.8M floats
  float* t3    = hsum1 + 12800000;      // 3.2M floats
  float* t4    = hsum1 + 16000000;      // 3.2M floats

  // zero the stat accumulators (d_ws is poisoned, and must be reset every call)
  zero_f32<<<2, 256, 0, stream>>>(stats, 320);

  // ---- layer 1 ----
  // hsum1 = x  (GIN eps=0: h = x + sum_neighbors)
  copy_f4<<<(N_NODES * F_IN / 4 + 255) / 256, 256, 0, stream>>>(
      (float4*)hsum1, (const float4*)x, N_NODES * F_IN / 4);
  // hsum1[dst] += x[src]
  {
    int total = N_EDGES * F_IN;  // 819,200,000 < 2^31
    scatter_add<<<(total + 255) / 256, 256, 0, stream>>>(x, srcIdx, dstIdx, hsum1,
                                                         total, F_IN);
  }
  // t1 = relu(hsum1 @ W1a + b1a)
  {
    int tiles = (N_NODES / 16) * (H1 / 16);  // 50,000
    gemm_wmma_f32<<<(tiles + 7) / 8, 256, 0, stream>>>(
        hsum1, W1a, b1a, t1, N_NODES, F_IN, H1, 1, nullptr, nullptr);
  }
  // t2 = t1 @ W1b + b1b   (+ accumulate BN batch stats)
  {
    int tiles = (N_NODES / 16) * (H1 / 16);
    gemm_wmma_f32<<<(tiles + 7) / 8, 256, 0, stream>>>(
        t1, W1b, b1b, t2, N_NODES, H1, H1, 0, ssum1, ssq1);
  }
  // h1 = relu(BN(t2))
  bn_relu<<<(N_NODES * H1 + 255) / 256, 256, 0, stream>>>(
      t2, ssum1, ssq1, g1, be1, h1, N_NODES * H1, H1, 1.0f / (float)N_NODES);

  // ---- layer 2 ----
  copy_f4<<<(N_NODES * H1 / 4 + 255) / 256, 256, 0, stream>>>(
      (float4*)hsum2, (const float4*)h1, N_NODES * H1 / 4);
  {
    int total = N_EDGES * H1;  // 409,600,000
    scatter_add<<<(total + 255) / 256, 256, 0, stream>>>(h1, srcIdx, dstIdx, hsum2,
                                                         total, H1);
  }
  {
    int tiles = (N_NODES / 16) * (H2 / 16);  // 12,500
    gemm_wmma_f32<<<(tiles + 7) / 8, 256, 0, stream>>>(
        hsum2, W2a, b2a, t3, N_NODES, H1, H2, 1, nullptr, nullptr);
    gemm_wmma_f32<<<(tiles + 7) / 8, 256, 0, stream>>>(
        t3, W2b, b2b, t4, N_NODES, H2, H2, 0, ssum2, ssq2);
  }
  bn_relu<<<(N_NODES * H2 + 255) / 256, 256, 0, stream>>>(
      t4, ssum2, ssq2, g2, be2, (float*)d_out, N_NODES * H2, H2,
      1.0f / (float)N_NODES);
}